// subspace_linop_old_13623636263108
// MI455X (gfx1250) — compile-verified
//
#include <hip/hip_runtime.h>

// Problem constants (from reference): nsub=5, ntr=100, grid 320x320.
#define NTR       100
#define GRID_Y    320
#define NBINS     102400          // 320*320
#define NROWS     50              // 2 (re/im) * 5*5 subspace pairs
#define NROWS_PAD 64              // 4 WMMA M-tiles
#define KSTEPS    25              // 100 / 4 (K per WMMA = 4)

typedef __attribute__((ext_vector_type(2))) float v2f;
typedef __attribute__((ext_vector_type(8))) float v8f;

// ---------------- zero the D accumulator (100 x 102400) ----------------
__global__ void zero_ws(float* __restrict__ p, int n) {
    int i = blockIdx.x * blockDim.x + threadIdx.x;
    if (i < n) p[i] = 0.0f;
}

// ---------------- build W (64 x 100, rows >= 50 zero-padded) ----------------
// row < 25 : w_re[a,b,t] = pr[a,t]*pr[b,t] + pi[a,t]*pi[b,t]
// 25..49   : w_im[a,b,t] = pr[a,t]*pi[b,t] - pi[a,t]*pr[b,t]
__global__ void build_w(const float* __restrict__ pr, const float* __restrict__ pi,
                        float* __restrict__ W) {
    int i = blockIdx.x * blockDim.x + threadIdx.x;
    if (i >= NROWS_PAD * NTR) return;
    int row = i / NTR, t = i % NTR;
    float v = 0.0f;
    if (row < NROWS) {
        int reim = row / 25;
        int rest = row % 25;
        int a = rest / 5, b = rest % 5;
        float pra = pr[a * NTR + t], prb = pr[b * NTR + t];
        float pia = pi[a * NTR + t], pib = pi[b * NTR + t];
        v = (reim == 0) ? (pra * prb + pia * pib)
                        : (pra * pib - pia * prb);
    }
    W[row * NTR + t] = v;
}

// ---------------- scatter dcf into D[t, flat] (1 atomic per point) ----------------
__global__ void scatter_dcf(const float* __restrict__ dcf, const int* __restrict__ trj,
                            float* __restrict__ D, int npts) {
    int k = blockIdx.x * blockDim.x + threadIdx.x;
    if (k >= npts) return;
    float v = dcf[k];
    int x = trj[2 * k + 0];
    int y = trj[2 * k + 1];
    int t = k % NTR;                       // ntr is the fastest-varying dcf dim
    int flat = x * GRID_Y + y;
    unsafeAtomicAdd(&D[(size_t)t * NBINS + flat], v);  // global_atomic_add_f32
}

// ---------------- WMMA GEMM: (64x100) x (100x102400) -> (50x102400) ----------------
// One wave per block computes a 64-row x 16-col output strip.
// V_WMMA_F32_16X16X4_F32 operand layout (ISA 7.12.2):
//   A 16x4: lanes 0-15 -> M=lane, K={0,1}; lanes 16-31 -> M=lane-16, K={2,3}
//   B 4x16: lanes 0-15 -> N=lane, K={0,1}; lanes 16-31 -> N=lane-16, K={2,3}
//   C/D:    VGPR j -> M = j + 8*(lane>=16), N = lane&15
__global__ __launch_bounds__(32) void gemm_wmma(const float* __restrict__ W,  // 64 x 100
                                                const float* __restrict__ D,  // 100 x 102400
                                                float* __restrict__ out) {    // 50 x 102400
    const int col0 = blockIdx.x * 16;
    const int lane = threadIdx.x;
    const int half = lane >> 4;      // 0: K pair {0,1}, 1: K pair {2,3}
    const int l16  = lane & 15;

    v8f acc[4] = {v8f{}, v8f{}, v8f{}, v8f{}};

    for (int kk = 0; kk < KSTEPS; ++kk) {
        const int k0 = kk * 4 + half * 2;

        // B operand: two consecutive K-rows of D at this lane's column
        v2f b;
        b.x = D[(size_t)(k0 + 0) * NBINS + col0 + l16];
        b.y = D[(size_t)(k0 + 1) * NBINS + col0 + l16];

        // A operands: 4 M-tiles of W (zero-padded rows -> no guard needed)
        #pragma unroll
        for (int m = 0; m < 4; ++m) {
            const int row = m * 16 + l16;
            v2f a;
            a.x = W[row * NTR + k0 + 0];
            a.y = W[row * NTR + k0 + 1];
            acc[m] = __builtin_amdgcn_wmma_f32_16x16x4_f32(
                /*neg_a=*/false, a, /*neg_b=*/false, b,
                /*c_mod=*/(short)0, acc[m],
                /*reuse_a=*/false, /*reuse_b=*/false);
        }
    }

    // Store: guard the 50 real output rows; column strip fully covered.
    const int col = col0 + l16;
    #pragma unroll
    for (int m = 0; m < 4; ++m) {
        #pragma unroll
        for (int j = 0; j < 8; ++j) {
            const int row = m * 16 + half * 8 + j;
            if (row < NROWS)
                out[(size_t)row * NBINS + col] = acc[m][j];
        }
    }
}

extern "C" void kernel_launch(void* const* d_in, const int* in_sizes, int n_in,
                              void* d_out, int out_size, void* d_ws, size_t ws_size,
                              hipStream_t stream) {
    const float* phi_re = (const float*)d_in[0];   // (5,100)
    const float* phi_im = (const float*)d_in[1];   // (5,100)
    const float* dcf    = (const float*)d_in[2];   // (1,512,48,100)
    const int*   trj    = (const int*)d_in[3];     // (1,512,48,100,2)
    float* out = (float*)d_out;                    // (2,1,5,5,320,320)

    float* D = (float*)d_ws;                       // 100 * 102400 f32 = 40.96 MB
    float* W = D + (size_t)NTR * NBINS;            // 64 * 100 f32

    const int npts = in_sizes[2];                  // 2,457,600
    const int nD = NTR * NBINS;

    zero_ws<<<(nD + 255) / 256, 256, 0, stream>>>(D, nD);
    build_w<<<(NROWS_PAD * NTR + 255) / 256, 256, 0, stream>>>(phi_re, phi_im, W);
    scatter_dcf<<<(npts + 255) / 256, 256, 0, stream>>>(dcf, trj, D, npts);
    gemm_wmma<<<NBINS / 16, 32, 0, stream>>>(W, D, out);
}